// GNNLinkPredictor_16192026706661
// MI455X (gfx1250) — compile-verified
//
#include <hip/hip_runtime.h>
#include <hip/hip_bf16.h>

typedef __attribute__((ext_vector_type(2))) float v2f;
typedef __attribute__((ext_vector_type(8))) float v8f;

// ---------------------------------------------------------------------------
// Utility: zero a float region (grid-stride; d_ws is poisoned 0xAA by harness)
// ---------------------------------------------------------------------------
__global__ void zero_f32(float* __restrict__ p, long long n) {
  long long i = (long long)blockIdx.x * blockDim.x + threadIdx.x;
  long long stride = (long long)gridDim.x * blockDim.x;
  for (; i < n; i += stride) p[i] = 0.0f;
}

// ---------------------------------------------------------------------------
// Degree count: cnt[dst[e]] += 1 ; then cnt -> 1/max(cnt,1)
// ---------------------------------------------------------------------------
__global__ void count_deg(const int* __restrict__ dst, float* __restrict__ cnt, int nedges) {
  int e = blockIdx.x * blockDim.x + threadIdx.x;
  if (e < nedges) atomicAdd(&cnt[dst[e]], 1.0f);
}

__global__ void invert_deg(float* __restrict__ cnt, int n) {
  int i = blockIdx.x * blockDim.x + threadIdx.x;
  if (i < n) cnt[i] = 1.0f / fmaxf(cnt[i], 1.0f);
}

// ---------------------------------------------------------------------------
// Edge scatter-add: agg[dst][*] += x[src][*]. C channels, 4 per thread.
// ---------------------------------------------------------------------------
template <int C>
__global__ void scatter_add(const float* __restrict__ x,
                            const int* __restrict__ src,
                            const int* __restrict__ dst,
                            float* __restrict__ agg, int nedges) {
  constexpr int G = C / 4;  // power of two (16 or 32)
  long long tid = (long long)blockIdx.x * blockDim.x + threadIdx.x;
  long long total = (long long)nedges * G;
  if (tid >= total) return;
  int e = (int)(tid / G);
  int g = (int)(tid % G);
  int s = src[e];
  int d = dst[e];
  const float4 v = ((const float4*)(x + (size_t)s * C))[g];
  float* a = agg + (size_t)d * C + (size_t)g * 4;
  atomicAdd(a + 0, v.x);
  atomicAdd(a + 1, v.y);
  atomicAdd(a + 2, v.z);
  atomicAdd(a + 3, v.w);
}

// ---------------------------------------------------------------------------
// Fused SAGE layer:  out = act( (agg .* inv) @ Wl + xin @ Wr + bias )
//
// V_WMMA_F32_16X16X4_F32 fragments (wave32):
//   A 16x4 : lane -> M = lane&15 ; v2f = A[M][k+2*kh], A[M][k+2*kh+1], kh=lane>>4
//   B 4x16 : lane -> N = lane&15 ; v2f = B[k+2*kh][N], B[k+2*kh+1][N]
//   C/D    : vgpr r, lane -> D[8*kh + r][lane&15]
//
// Layout: one wave per output-channel tile (NT waves/block). Each wave keeps
// ALL of its B fragments (both weight matrices) in registers across the whole
// M loop -- 64 VGPRs (layer1) / 128 VGPRs (layer2), no spills. The 16-row A
// tile is block-shared: cooperatively staged to LDS with the mean division
// folded in, +4-float row pad for conflict-free ds_load_b64.
// Inner loop: clause of 16 ds_load_b64 (A) -> 16 back-to-back v_wmma.
// ---------------------------------------------------------------------------
template <int CI, int CO, bool RELU>
__global__ __launch_bounds__((CO / 16) * 32) void sage_gemm(
    const float* __restrict__ agg, const float* __restrict__ invd,
    const float* __restrict__ xin,
    const float* __restrict__ Wl, const float* __restrict__ Wr,
    const float* __restrict__ bias,
    float* __restrict__ out, int ntiles) {
  constexpr int KS = CI / 4;      // WMMA K-steps per matrix
  constexpr int LROW = CI + 4;    // padded LDS row stride (floats)
  constexpr int CHUNK = 16;       // K-steps staged per register chunk

  __shared__ float aggS[16 * LROW];  // pre-scaled mean tile
  __shared__ float xS[16 * LROW];    // self-feature tile

  const int lane = threadIdx.x & 31;
  const int mr = lane & 15;   // A row / B,C,D column within tile
  const int kh = lane >> 4;   // K-half selector
  const int nt = threadIdx.x >> 5;  // wave id == output-channel tile
  const int col = nt * 16 + mr;

  // Register-resident B fragments for this wave's column tile (both matrices).
  v2f bl[KS], br[KS];
#pragma unroll
  for (int ks = 0; ks < KS; ++ks) {
    const int k = 4 * ks + 2 * kh;
    bl[ks].x = Wl[(size_t)k * CO + col];
    bl[ks].y = Wl[(size_t)(k + 1) * CO + col];
    br[ks].x = Wr[(size_t)k * CO + col];
    br[ks].y = Wr[(size_t)(k + 1) * CO + col];
  }
  const float bb = bias[col];

  for (int mt = blockIdx.x; mt < ntiles; mt += gridDim.x) {
    const int m0 = mt * 16;

    __syncthreads();  // previous iteration's readers done before restage
    for (int i = threadIdx.x; i < 16 * CI; i += blockDim.x) {
      const int r = i / CI, c = i - r * CI;
      const float sc = invd[m0 + r];
      aggS[r * LROW + c] = agg[(size_t)(m0 + r) * CI + c] * sc;
      xS[r * LROW + c] = xin[(size_t)(m0 + r) * CI + c];
    }
    __syncthreads();

    v8f acc = (v8f)0.0f;
    const float* aRow0 = &aggS[mr * LROW + 2 * kh];
    const float* aRow1 = &xS[mr * LROW + 2 * kh];

    // pass 1: mean @ Wl
#pragma unroll
    for (int kc = 0; kc < KS; kc += CHUNK) {
      v2f areg[CHUNK];
#pragma unroll
      for (int j = 0; j < CHUNK; ++j)  // ds_load_b64 clause
        areg[j] = *(const v2f*)(aRow0 + 4 * (kc + j));
#pragma unroll
      for (int j = 0; j < CHUNK; ++j)
        acc = __builtin_amdgcn_wmma_f32_16x16x4_f32(
            false, areg[j], false, bl[kc + j], (short)0, acc, false, false);
    }
    // pass 2: x @ Wr
#pragma unroll
    for (int kc = 0; kc < KS; kc += CHUNK) {
      v2f areg[CHUNK];
#pragma unroll
      for (int j = 0; j < CHUNK; ++j)
        areg[j] = *(const v2f*)(aRow1 + 4 * (kc + j));
#pragma unroll
      for (int j = 0; j < CHUNK; ++j)
        acc = __builtin_amdgcn_wmma_f32_16x16x4_f32(
            false, areg[j], false, br[kc + j], (short)0, acc, false, false);
    }

    // writeback: bias + optional ReLU
#pragma unroll
    for (int r8 = 0; r8 < 8; ++r8) {
      float v = acc[r8] + bb;
      if (RELU) v = fmaxf(v, 0.0f);
      out[(size_t)(m0 + kh * 8 + r8) * CO + col] = v;
    }
  }
}

// ---------------------------------------------------------------------------
// Decode: out[p] = dot(z[sl[p]], z[dl[p]]), 64-dim, float4 loads
// ---------------------------------------------------------------------------
__global__ void decode_dot(const float* __restrict__ z,
                           const int* __restrict__ sl,
                           const int* __restrict__ dl,
                           float* __restrict__ out, int np) {
  int p = blockIdx.x * blockDim.x + threadIdx.x;
  if (p >= np) return;
  const float4* a = (const float4*)(z + (size_t)sl[p] * 64);
  const float4* b = (const float4*)(z + (size_t)dl[p] * 64);
  float s = 0.0f;
#pragma unroll
  for (int i = 0; i < 16; ++i) {
    float4 u = a[i];
    float4 v = b[i];
    s += u.x * v.x + u.y * v.y + u.z * v.z + u.w * v.w;
  }
  out[p] = s;
}

// ---------------------------------------------------------------------------
// Launcher
// inputs: 0:x 1:W_l1 2:W_r1 3:b1 4:W_l2 5:W_r2 6:b2 7:edge_index 8:edge_label_index
// ---------------------------------------------------------------------------
extern "C" void kernel_launch(void* const* d_in, const int* in_sizes, int n_in,
                              void* d_out, int out_size, void* d_ws, size_t ws_size,
                              hipStream_t stream) {
  constexpr int IN_CH = 64, HID_CH = 128, OUT_CH = 64;

  const float* x   = (const float*)d_in[0];
  const float* Wl1 = (const float*)d_in[1];
  const float* Wr1 = (const float*)d_in[2];
  const float* b1  = (const float*)d_in[3];
  const float* Wl2 = (const float*)d_in[4];
  const float* Wr2 = (const float*)d_in[5];
  const float* b2  = (const float*)d_in[6];
  const int* ei    = (const int*)d_in[7];
  const int* eli   = (const int*)d_in[8];

  const int nnodes = in_sizes[0] / IN_CH;
  const int nedges = in_sizes[7] / 2;
  const int nlabel = in_sizes[8] / 2;
  const int ntiles = nnodes / 16;  // 6250, exact

  const int* esrc = ei;
  const int* edst = ei + nedges;
  const int* lsrc = eli;
  const int* ldst = eli + nlabel;

  // workspace layout (floats): inv | agg(128ch, reused) | h(128ch) | z(64ch)
  float* ws  = (float*)d_ws;
  float* inv = ws;
  float* agg = inv + nnodes;
  float* h   = agg + (size_t)nnodes * HID_CH;
  float* z   = h + (size_t)nnodes * HID_CH;

  const int B = 256;

  // zero inv + first-64ch of agg in one pass (adjacent)
  zero_f32<<<2048, B, 0, stream>>>(inv, (long long)nnodes * (1 + IN_CH));

  // degree -> inverse degree
  count_deg<<<(nedges + B - 1) / B, B, 0, stream>>>(edst, inv, nedges);
  invert_deg<<<(nnodes + B - 1) / B, B, 0, stream>>>(inv, nnodes);

  // layer 1: aggregate x, fused GEMM + ReLU (8 waves/block = 8 col-tiles)
  {
    long long nth = (long long)nedges * (IN_CH / 4);
    scatter_add<IN_CH><<<(int)((nth + B - 1) / B), B, 0, stream>>>(x, esrc, edst, agg, nedges);
    sage_gemm<IN_CH, HID_CH, true><<<512, (HID_CH / 16) * 32, 0, stream>>>(
        agg, inv, x, Wl1, Wr1, b1, h, ntiles);
  }

  // layer 2: re-zero agg (now 128ch), aggregate h, GEMM (4 waves/block)
  zero_f32<<<2048, B, 0, stream>>>(agg, (long long)nnodes * HID_CH);
  {
    long long nth = (long long)nedges * (HID_CH / 4);
    scatter_add<HID_CH><<<(int)((nth + B - 1) / B), B, 0, stream>>>(h, esrc, edst, agg, nedges);
    sage_gemm<HID_CH, OUT_CH, false><<<1024, (OUT_CH / 16) * 32, 0, stream>>>(
        agg, inv, h, Wl2, Wr2, b2, z, ntiles);
  }

  // link decode
  decode_dot<<<(nlabel + B - 1) / B, B, 0, stream>>>(z, lsrc, ldst, (float*)d_out, nlabel);
}